// Encoder_29901562315294
// MI455X (gfx1250) — compile-verified
//
#include <hip/hip_runtime.h>
#include <math.h>

// ---------------------------------------------------------------------------
// MI455X (gfx1250) implementation of the S4 encoder reference.
// f32 WMMA (16x16x4) for all GEMM-shaped work (memory-bound -> keep f32),
// LDS-staged 64x64 tiles via async global->LDS copies,
// LDS-resident radix-2 FFT for the SSM long convolution.
// ---------------------------------------------------------------------------

typedef float v2f __attribute__((ext_vector_type(2)));
typedef float v8f __attribute__((ext_vector_type(8)));

#if defined(__has_builtin)
#if __has_builtin(__builtin_amdgcn_global_load_async_to_lds_b32) && \
    __has_builtin(__builtin_amdgcn_s_wait_asynccnt)
#define HAVE_ASYNC_LDS 1
#endif
#endif

#ifdef HAVE_ASYNC_LDS
typedef __attribute__((address_space(1))) int gas_int;
typedef __attribute__((address_space(3))) int las_int;
#endif

__device__ __forceinline__ float gelu_tanh(float v) {
  float v3 = v * v * v;
  return 0.5f * v * (1.0f + tanhf(0.7978845608028654f * (v + 0.044715f * v3)));
}
__device__ __forceinline__ float sigmoidf(float v) { return 1.0f / (1.0f + __expf(-v)); }
__device__ __forceinline__ unsigned bitrev(unsigned v, int bits) { return __brev(v) >> (32 - bits); }

// ---------------------------------------------------------------------------
// BatchNorm (eval) + embedding linear: F=12 -> H=128. One block per row.
// ---------------------------------------------------------------------------
__global__ void bn_embed_kernel(const float* __restrict__ x,
                                const float* __restrict__ g, const float* __restrict__ bb,
                                const float* __restrict__ mean, const float* __restrict__ var,
                                const float* __restrict__ ew, const float* __restrict__ eb,
                                float* __restrict__ out, int F, int H) {
  int row = blockIdx.x;
  int t = threadIdx.x;
  __shared__ float xn[32];
  if (t < F) {
    float v = x[(size_t)row * F + t];
    xn[t] = (v - mean[t]) * rsqrtf(var[t] + 1e-5f) * g[t] + bb[t];
  }
  __syncthreads();
  if (t < H) {
    float acc = eb[t];
    const float* wr = ew + (size_t)t * F;
    for (int f = 0; f < F; ++f) acc += xn[f] * wr[f];
    out[(size_t)row * H + t] = acc;
  }
}

// ---------------------------------------------------------------------------
// Row-wise LayerNorm. One block (256 threads) per row, C in {128,256,512}.
// ---------------------------------------------------------------------------
__global__ void layernorm_kernel(const float* __restrict__ in,
                                 const float* __restrict__ g, const float* __restrict__ bb,
                                 float* __restrict__ out, int C) {
  int row = blockIdx.x;
  const float* r = in + (size_t)row * C;
  __shared__ float red[256];
  float s = 0.f;
  for (int c = threadIdx.x; c < C; c += blockDim.x) s += r[c];
  red[threadIdx.x] = s; __syncthreads();
  for (int o = blockDim.x >> 1; o > 0; o >>= 1) {
    if (threadIdx.x < o) red[threadIdx.x] += red[threadIdx.x + o];
    __syncthreads();
  }
  float m = red[0] / (float)C;
  __syncthreads();
  float v = 0.f;
  for (int c = threadIdx.x; c < C; c += blockDim.x) { float d = r[c] - m; v += d * d; }
  red[threadIdx.x] = v; __syncthreads();
  for (int o = blockDim.x >> 1; o > 0; o >>= 1) {
    if (threadIdx.x < o) red[threadIdx.x] += red[threadIdx.x + o];
    __syncthreads();
  }
  float inv = rsqrtf(red[0] / (float)C + 1e-5f);
  for (int c = threadIdx.x; c < C; c += blockDim.x)
    out[(size_t)row * C + c] = (r[c] - m) * inv * g[c] + bb[c];
}

// ---------------------------------------------------------------------------
// SSM coefficients: dtA = dt*A, Ceff = (C_re+iC_im)*(exp(dtA)-1)/A.
// coef layout: [0]=dtA_re, [1]=dtA_im, [2,3]=Ceff(k=0) re/im, [4,5]=Ceff(k=1).
// ---------------------------------------------------------------------------
__global__ void ssm_coef_kernel(const float* __restrict__ log_dt,
                                const float* __restrict__ log_A_real,
                                const float* __restrict__ A_imag,
                                const float* __restrict__ C_re, const float* __restrict__ C_im,
                                float* __restrict__ coef, int Hc, int N) {
  int i = blockIdx.x * blockDim.x + threadIdx.x;
  int tot = Hc * N;
  if (i >= tot) return;
  int h = i / N;
  float dt = __expf(log_dt[h]);
  float Ar = -__expf(log_A_real[i]);
  float Ai = A_imag[i];
  float dr = dt * Ar, di = dt * Ai;
  float er = __expf(dr);
  float sn, cs; __sincosf(di, &sn, &cs);
  float num_r = er * cs - 1.0f, num_i = er * sn;
  float den = Ar * Ar + Ai * Ai;
  float fr = (num_r * Ar + num_i * Ai) / den;
  float fi = (num_i * Ar - num_r * Ai) / den;
  coef[0 * tot + i] = dr;
  coef[1 * tot + i] = di;
  for (int k = 0; k < 2; ++k) {
    float cr = C_re[k * tot + i], ci = C_im[k * tot + i];
    coef[(2 + 2 * k) * tot + i] = cr * fr - ci * fi;
    coef[(3 + 2 * k) * tot + i] = cr * fi + ci * fr;
  }
}

// ---------------------------------------------------------------------------
// Bidirectional time-domain kernel kpad(Hc, 2Lx):
//   l2<Lx -> K0[h,l2]; l2>=Lx -> K1[h,2Lx-1-l2];  K = 2*Re(sum_n Ceff*exp(dtA*l))
// ---------------------------------------------------------------------------
__global__ void ssm_kernel_build(const float* __restrict__ coef, float* __restrict__ kt,
                                 int Hc, int N, int Lx) {
  int L2 = 2 * Lx;
  long long idx = (long long)blockIdx.x * blockDim.x + threadIdx.x;
  if (idx >= (long long)Hc * L2) return;
  int h = (int)(idx / L2), l2 = (int)(idx % L2);
  int k, l;
  if (l2 < Lx) { k = 0; l = l2; } else { k = 1; l = 2 * Lx - 1 - l2; }
  int tot = Hc * N;
  const float* dr = coef + 0 * tot + h * N;
  const float* di = coef + 1 * tot + h * N;
  const float* cr = coef + (2 + 2 * k) * tot + h * N;
  const float* ci = coef + (3 + 2 * k) * tot + h * N;
  float fl = (float)l;
  float acc = 0.f;
  for (int n = 0; n < N; ++n) {
    float er = __expf(dr[n] * fl);
    float sn, cs; __sincosf(di[n] * fl, &sn, &cs);
    acc += er * (cr[n] * cs - ci[n] * sn);
  }
  kt[idx] = 2.0f * acc;
}

// ---------------------------------------------------------------------------
// In-LDS radix-2 DIT FFT. Input must be bit-reversed; output natural order.
// ---------------------------------------------------------------------------
__device__ __forceinline__ void fft_lds(float2* s, int n, int logn, float sign) {
  for (int st = 1; st <= logn; ++st) {
    int half = 1 << (st - 1);
    float ang = sign * 6.283185307179586f / (float)(half << 1);
    for (int t = threadIdx.x; t < (n >> 1); t += blockDim.x) {
      int j = t & (half - 1);
      int base = ((t >> (st - 1)) << st) + j;
      float wc, wsn; __sincosf(ang * (float)j, &wsn, &wc);
      float2 a = s[base], b = s[base + half];
      float br = b.x * wc - b.y * wsn;
      float bi = b.x * wsn + b.y * wc;
      s[base].x = a.x + br; s[base].y = a.y + bi;
      s[base + half].x = a.x - br; s[base + half].y = a.y - bi;
    }
    __syncthreads();
  }
}

// Forward FFT of the SSM kernel: one workgroup per channel h.
__global__ void kernel_fft_kernel(const float* __restrict__ kt, float2* __restrict__ kf,
                                  int Lx, int logn) {
  extern __shared__ float2 sm[];
  int n = 2 * Lx;
  int h = blockIdx.x;
  const float* src = kt + (size_t)h * n;
  for (int i = threadIdx.x; i < n; i += blockDim.x) {
    float2 v; v.x = src[i]; v.y = 0.f;
    sm[bitrev((unsigned)i, logn)] = v;
  }
  __syncthreads();
  fft_lds(sm, n, logn, -1.0f);
  float2* dst = kf + (size_t)h * n;
  for (int i = threadIdx.x; i < n; i += blockDim.x) dst[i] = sm[i];
}

// FFT conv + D-skip + GELU fused: one workgroup per (b,h) channel.
__global__ void fft_conv_kernel(const float* __restrict__ u, const float2* __restrict__ kf,
                                const float* __restrict__ D, float* __restrict__ y,
                                int Hc, int Lx, int logn) {
  extern __shared__ float2 sm[];
  int n = 2 * Lx;
  int bh = blockIdx.x;
  int b = bh / Hc, h = bh % Hc;
  const float* ub = u + (size_t)b * Lx * Hc + h;
  for (int i = threadIdx.x; i < n; i += blockDim.x) {
    float2 v;
    v.x = (i < Lx) ? ub[(size_t)i * Hc] : 0.f;
    v.y = 0.f;
    sm[bitrev((unsigned)i, logn)] = v;
  }
  __syncthreads();
  fft_lds(sm, n, logn, -1.0f);
  const float2* kfh = kf + (size_t)h * n;
  for (int i = threadIdx.x; i < n; i += blockDim.x) {
    float2 a = sm[i], k2 = kfh[i];
    float2 r; r.x = a.x * k2.x - a.y * k2.y; r.y = a.x * k2.y + a.y * k2.x;
    sm[i] = r;
  }
  __syncthreads();
  // bit-reverse permute (pairwise swap) before inverse FFT
  for (int i = threadIdx.x; i < n; i += blockDim.x) {
    int r = (int)bitrev((unsigned)i, logn);
    if (r > i) { float2 t = sm[i]; sm[i] = sm[r]; sm[r] = t; }
  }
  __syncthreads();
  fft_lds(sm, n, logn, 1.0f);
  float scale = 1.0f / (float)n;
  float dh = D[h];
  for (int i = threadIdx.x; i < Lx; i += blockDim.x) {
    float uv = ub[(size_t)i * Hc];
    float yy = sm[i].x * scale + uv * dh;
    y[((size_t)b * Lx + i) * Hc + h] = gelu_tanh(yy);
  }
}

// ---------------------------------------------------------------------------
// WMMA GEMM with LDS-staged 64x64 block tiles (async global->LDS copies).
// out[M,N] = epilogue(A[M,K](lda) @ W[N,K]^T + bias)
// mode 0: store; mode 1: GELU then store; mode 2: accumulate (residual).
// 128 threads = 4 waves; wave w owns a 64x16 column stripe (4 accumulators).
// Requires M%64==0, N%64==0, K%32==0 (true for all shapes in this model).
// ---------------------------------------------------------------------------
#define BM 64
#define BN 64
#define KT 32
#define KTP 34   // padded LDS row stride (floats) to dodge bank conflicts

__global__ __launch_bounds__(128)
void wmma_gemm_kernel(const float* __restrict__ A, int lda,
                      const float* __restrict__ W, const float* __restrict__ bias,
                      float* __restrict__ out, int ldo,
                      int M, int Nn, int K, int mode) {
  __shared__ float smA[BM * KTP];
  __shared__ float smW[BN * KTP];
  int tid = threadIdx.x;
  int lane = tid & 31;
  int wave = tid >> 5;
  int blockM = blockIdx.x * BM;
  int blockN = blockIdx.y * BN;

  v8f acc[4];
#pragma unroll
  for (int i = 0; i < 4; ++i) acc[i] = (v8f){0.f,0.f,0.f,0.f,0.f,0.f,0.f,0.f};

  int lane15 = lane & 15;
  int khalf = (lane >> 4) << 1;            // lanes 0-15: K 0,1 ; 16-31: K 2,3
  int nlocal = wave * 16 + lane15;         // this wave's column inside the tile

  for (int k0 = 0; k0 < K; k0 += KT) {
    // ---- stage A(64xKT) and W(64xKT) into LDS ----
#pragma unroll
    for (int i = tid; i < BM * KT; i += 128) {
      int r = i >> 5, c = i & (KT - 1);
      const float* src = A + (size_t)(blockM + r) * lda + k0 + c;
#ifdef HAVE_ASYNC_LDS
      __builtin_amdgcn_global_load_async_to_lds_b32(
          (gas_int*)(const void*)src,
          (las_int*)(void*)&smA[r * KTP + c], 0, 0);
#else
      smA[r * KTP + c] = *src;
#endif
    }
#pragma unroll
    for (int i = tid; i < BN * KT; i += 128) {
      int r = i >> 5, c = i & (KT - 1);
      const float* src = W + (size_t)(blockN + r) * K + k0 + c;
#ifdef HAVE_ASYNC_LDS
      __builtin_amdgcn_global_load_async_to_lds_b32(
          (gas_int*)(const void*)src,
          (las_int*)(void*)&smW[r * KTP + c], 0, 0);
#else
      smW[r * KTP + c] = *src;
#endif
    }
    // prefetch next K-tile while this one is in flight
    if (k0 + KT < K) {
      __builtin_prefetch(&A[(size_t)(blockM + (tid & 63)) * lda + k0 + KT], 0, 1);
      __builtin_prefetch(&W[(size_t)(blockN + (tid & 63)) * K + k0 + KT], 0, 1);
    }
#ifdef HAVE_ASYNC_LDS
    __builtin_amdgcn_s_wait_asynccnt(0);
#endif
    __syncthreads();

    // ---- 8 x (4 subtile) WMMA steps out of LDS ----
#pragma unroll
    for (int kk = 0; kk < KT; kk += 4) {
      v2f b;
      b.x = smW[nlocal * KTP + kk + khalf];
      b.y = smW[nlocal * KTP + kk + khalf + 1];
#pragma unroll
      for (int mi = 0; mi < 4; ++mi) {
        v2f a;
        int mrow = mi * 16 + lane15;
        a.x = smA[mrow * KTP + kk + khalf];
        a.y = smA[mrow * KTP + kk + khalf + 1];
        acc[mi] = __builtin_amdgcn_wmma_f32_16x16x4_f32(false, a, false, b,
                                                        (short)0, acc[mi], false, false);
      }
    }
    __syncthreads();
  }

  // ---- epilogue ----
  int mhi = (lane >> 4) * 8;
  float bval = bias[blockN + nlocal];
#pragma unroll
  for (int mi = 0; mi < 4; ++mi) {
#pragma unroll
    for (int r = 0; r < 8; ++r) {
      int mr = blockM + mi * 16 + mhi + r;
      size_t o = (size_t)mr * ldo + blockN + nlocal;
      float v = acc[mi][r] + bval;
      if (mode == 1) v = gelu_tanh(v);
      if (mode == 2) v += out[o];
      out[o] = v;
    }
  }
}

// GLU + residual: x[m,c] += z[m,c] * sigmoid(z[m,Hc+c])
__global__ void glu_residual_kernel(float* __restrict__ x, const float* __restrict__ z,
                                    long long total, int Hc) {
  long long i = (long long)blockIdx.x * blockDim.x + threadIdx.x;
  if (i >= total) return;
  long long m = i / Hc; int c = (int)(i % Hc);
  const float* zr = z + m * (long long)(2 * Hc);
  x[i] += zr[c] * sigmoidf(zr[Hc + c]);
}

// Repack conv weight (O,C,2) row-major -> (O, 2C) with layout [h*C + c]
__global__ void conv_repack_kernel(const float* __restrict__ w, float* __restrict__ wr,
                                   int O, int C) {
  int i = blockIdx.x * blockDim.x + threadIdx.x;
  if (i >= O * C * 2) return;
  int o = i / (C * 2); int rem = i % (C * 2); int h = rem / C; int c = rem % C;
  wr[(size_t)o * (2 * C) + h * C + c] = w[((size_t)o * C + c) * 2 + h];
}

// Head (C->NC) + nearest-exact upsample Lin -> Lout.
__global__ void head_upsample_kernel(const float* __restrict__ x,
                                     const float* __restrict__ hw, const float* __restrict__ hb,
                                     float* __restrict__ out,
                                     int B, int Lout, int Lin, int C, int NCk) {
  int idx = blockIdx.x * blockDim.x + threadIdx.x;
  if (idx >= B * Lout * NCk) return;
  int k = idx % NCk;
  int l = (idx / NCk) % Lout;
  int b = idx / (NCk * Lout);
  int li = (int)floorf(((float)l + 0.5f) * (float)Lin / (float)Lout);
  li = min(max(li, 0), Lin - 1);
  const float* xr = x + ((size_t)b * Lin + li) * C;
  const float* wr = hw + (size_t)k * C;
  float acc = hb[k];
  for (int c = 0; c < C; ++c) acc += xr[c] * wr[c];
  out[idx] = acc;
}

// ---------------------------------------------------------------------------
// Host orchestration
// ---------------------------------------------------------------------------
struct S4Params {
  const float *ln1_g, *ln1_b, *ln2_g, *ln2_b, *log_dt, *log_A_real, *A_imag,
              *C_re, *C_im, *D, *w2, *b2, *w3, *b3, *wo, *bo;
};

static S4Params load_block(void* const* d_in, int base) {
  S4Params p;
  const float** f = (const float**)&p;
  for (int i = 0; i < 16; ++i) f[i] = (const float*)d_in[base + i];
  return p;
}

static void run_s4_block(const S4Params& p, float* X, float* U, float* Y, float* Z,
                         float* kt, float2* kf, float* coef,
                         int B, int Lx, int Hc, int N, int logn, hipStream_t stream) {
  size_t M = (size_t)B * Lx;
  size_t shm = (size_t)2 * Lx * sizeof(float2);
  // ln1 -> U
  layernorm_kernel<<<(unsigned)M, 256, 0, stream>>>(X, p.ln1_g, p.ln1_b, U, Hc);
  // SSM kernel: coef -> time kernel -> spectrum
  int cn = Hc * N;
  ssm_coef_kernel<<<(cn + 255) / 256, 256, 0, stream>>>(p.log_dt, p.log_A_real, p.A_imag,
                                                        p.C_re, p.C_im, coef, Hc, N);
  long long kn = (long long)Hc * 2 * Lx;
  ssm_kernel_build<<<(unsigned)((kn + 255) / 256), 256, 0, stream>>>(coef, kt, Hc, N, Lx);
  kernel_fft_kernel<<<Hc, 256, shm, stream>>>(kt, kf, Lx, logn);
  // FFT convolution + GELU(y + u*D) -> Y
  fft_conv_kernel<<<B * Hc, 256, shm, stream>>>(U, kf, p.D, Y, Hc, Lx, logn);
  // z = Y @ wo^T + bo  (Hc -> 2Hc), then GLU + residual into X
  dim3 g1((unsigned)(M / 64), (unsigned)((2 * Hc) / 64));
  wmma_gemm_kernel<<<g1, 128, 0, stream>>>(Y, Hc, p.wo, p.bo, Z, 2 * Hc, (int)M, 2 * Hc, Hc, 0);
  long long tot = (long long)M * Hc;
  glu_residual_kernel<<<(unsigned)((tot + 255) / 256), 256, 0, stream>>>(X, Z, tot, Hc);
  // MLP: ln2 -> U; GELU(U@w2^T+b2) -> Z; X += Z@w3^T+b3
  layernorm_kernel<<<(unsigned)M, 256, 0, stream>>>(X, p.ln2_g, p.ln2_b, U, Hc);
  wmma_gemm_kernel<<<g1, 128, 0, stream>>>(U, Hc, p.w2, p.b2, Z, 2 * Hc, (int)M, 2 * Hc, Hc, 1);
  dim3 g2((unsigned)(M / 64), (unsigned)(Hc / 64));
  wmma_gemm_kernel<<<g2, 128, 0, stream>>>(Z, 2 * Hc, p.w3, p.b3, X, Hc, (int)M, Hc, 2 * Hc, 2);
}

extern "C" void kernel_launch(void* const* d_in, const int* in_sizes, int n_in,
                              void* d_out, int out_size, void* d_ws, size_t ws_size,
                              hipStream_t stream) {
  (void)in_sizes; (void)n_in; (void)out_size; (void)ws_size;
  const int Bv = 16, Lv = 4096, Fv = 12, Hv = 128, Nv = 64, NCv = 8;
  const int H2 = 256, H4 = 512;

  // Allow up to 128 KiB dynamic LDS for the FFT kernels (stage-1 = 64 KiB).
  (void)hipFuncSetAttribute((const void*)kernel_fft_kernel,
                            hipFuncAttributeMaxDynamicSharedMemorySize, 1 << 17);
  (void)hipFuncSetAttribute((const void*)fft_conv_kernel,
                            hipFuncAttributeMaxDynamicSharedMemorySize, 1 << 17);

  char* ws = (char*)d_ws;
  const size_t MB = 1ull << 20;
  float*  bufA   = (float*)(ws + 0 * MB);     // 32 MiB activations
  float*  bufB   = (float*)(ws + 32 * MB);    // 32 MiB ln / downsample target
  float*  bufC   = (float*)(ws + 64 * MB);    // 32 MiB s4 conv output
  float*  bufZ   = (float*)(ws + 96 * MB);    // 64 MiB 2Hc-wide GEMM output
  float*  bufKT  = (float*)(ws + 160 * MB);   // 8 MiB time-domain kernel
  float2* bufKF  = (float2*)(ws + 168 * MB);  // 16 MiB kernel spectrum
  float*  bufCoef= (float*)(ws + 184 * MB);   // 2 MiB SSM coefficients
  float*  bufWR  = (float*)(ws + 186 * MB);   // 2 MiB repacked conv weights

  const float* xin  = (const float*)d_in[0];
  const float* bn_g = (const float*)d_in[1];
  const float* bn_b = (const float*)d_in[2];
  const float* bn_m = (const float*)d_in[3];
  const float* bn_v = (const float*)d_in[4];
  const float* emb_w = (const float*)d_in[5];
  const float* emb_b = (const float*)d_in[6];

  float* X = bufA; float* U = bufB; float* Y = bufC;

  // BatchNorm + embed
  bn_embed_kernel<<<Bv * Lv, Hv, 0, stream>>>(xin, bn_g, bn_b, bn_m, bn_v, emb_w, emb_b, X, Fv, Hv);

  // Stage 1: H=128, L=4096 (2L=8192, log2=13)
  for (int i = 0; i < 2; ++i) {
    S4Params p = load_block(d_in, 7 + i * 16);
    run_s4_block(p, X, U, Y, bufZ, bufKT, bufKF, bufCoef, Bv, Lv, Hv, Nv, 13, stream);
  }
  // Downsample 1: (B,4096,128) -> (B,2048,256); conv == GEMM with lda = 2C
  {
    const float* w = (const float*)d_in[103];
    const float* b = (const float*)d_in[104];
    int tot = H2 * Hv * 2;
    conv_repack_kernel<<<(tot + 255) / 256, 256, 0, stream>>>(w, bufWR, H2, Hv);
    size_t Md = (size_t)Bv * (Lv / 2);
    dim3 g((unsigned)(Md / 64), (unsigned)(H2 / 64));
    wmma_gemm_kernel<<<g, 128, 0, stream>>>(X, 2 * Hv, bufWR, b, U, H2, (int)Md, H2, 2 * Hv, 0);
    float* t = X; X = U; U = t;
  }
  // Stage 2: H=256, L=2048 (log2(2L)=12)
  for (int i = 0; i < 2; ++i) {
    S4Params p = load_block(d_in, 39 + i * 16);
    run_s4_block(p, X, U, Y, bufZ, bufKT, bufKF, bufCoef, Bv, Lv / 2, H2, Nv, 12, stream);
  }
  // Downsample 2: (B,2048,256) -> (B,1024,512)
  {
    const float* w = (const float*)d_in[105];
    const float* b = (const float*)d_in[106];
    int tot = H4 * H2 * 2;
    conv_repack_kernel<<<(tot + 255) / 256, 256, 0, stream>>>(w, bufWR, H4, H2);
    size_t Md = (size_t)Bv * (Lv / 4);
    dim3 g((unsigned)(Md / 64), (unsigned)(H4 / 64));
    wmma_gemm_kernel<<<g, 128, 0, stream>>>(X, 2 * H2, bufWR, b, U, H4, (int)Md, H4, 2 * H2, 0);
    float* t = X; X = U; U = t;
  }
  // Stage 3: H=512, L=1024 (log2(2L)=11)
  for (int i = 0; i < 2; ++i) {
    S4Params p = load_block(d_in, 71 + i * 16);
    run_s4_block(p, X, U, Y, bufZ, bufKT, bufKF, bufCoef, Bv, Lv / 4, H4, Nv, 11, stream);
  }
  // Head + nearest-exact upsample to L=4096
  {
    const float* hw = (const float*)d_in[107];
    const float* hb = (const float*)d_in[108];
    int tot = Bv * Lv * NCv;
    head_upsample_kernel<<<(tot + 255) / 256, 256, 0, stream>>>(X, hw, hb, (float*)d_out,
                                                                Bv, Lv, Lv / 4, H4, NCv);
  }
}